// RandLANetRes_89481348645716
// MI455X (gfx1250) — compile-verified
//
#include <hip/hip_runtime.h>
#include <hip/hip_bf16.h>

// ---------------------------------------------------------------------------
// RandLA-Net residual block on gfx1250 (wave32, WMMA f32 16x16x4).
// All dense math on V_WMMA_F32_16X16X4_F32 (exact f32, matches reference).
// Weights staged in LDS pre-packed so every B-fragment is one ds_load_b64.
// Segment-sum exploits dst = repeat(arange(Nq), 16): one wave == one target.
// ---------------------------------------------------------------------------

typedef __attribute__((ext_vector_type(2))) float v2f;
typedef __attribute__((ext_vector_type(8))) float v8f;

#define N_PTS   65536
#define N1_PTS  16384
#define N2_PTS  8192
#define KNN     16
#define INDIM   64
#define HID     32
#define OUTDIM  128

__device__ __forceinline__ v8f wmma_f32_4(v2f a, v2f b, v8f c) {
  // D = A(16x4) * B(4x16) + C(16x16), f32
  return __builtin_amdgcn_wmma_f32_16x16x4_f32(false, a, false, b,
                                               (short)0, c, false, false);
}

// Packed-weight index: Wp[(k>>1)*2N + c*2 + (k&1)] = W[k*N + c]
// => B-frag for k0 = kc*4+2*half, col = nt*16+n is the v2f at
//    Wp[(kc*2+half)*2N + col*2]   (8B aligned, conflict-free b64)

// ---------------------------------------------------------------------------
// GEMM + bias + ReLU:  out[M,N] = relu(A[M,K] @ W[K,N] + b[N])
// Block = 4 waves; weights packed in LDS once; each wave does RITER 16-row
// strips (A strip in LDS, stride K+4 => conflict-free fragment reads).
// ---------------------------------------------------------------------------
template <int K, int N, int RITER>
__global__ void gemm_bias_relu(const float* __restrict__ A,
                               const float* __restrict__ W,
                               const float* __restrict__ bias,
                               float* __restrict__ out) {
  constexpr int KP = K + 4;
  extern __shared__ float smem[];
  float* Wp = smem;                 // K*N packed
  float* bS = Wp + K * N;           // N
  float* Astrips = bS + N;          // 4 * 16 * KP

  const int tid  = threadIdx.x;
  const int lane = tid & 31;
  const int wv   = tid >> 5;

  for (int idx = tid; idx < K * N; idx += 128) {
    int k = idx / N, c = idx % N;
    Wp[(k >> 1) * (2 * N) + c * 2 + (k & 1)] = W[idx];
  }
  for (int idx = tid; idx < N; idx += 128) bS[idx] = bias[idx];
  __syncthreads();

  float* As = Astrips + wv * 16 * KP;
  const int m    = lane & 15;       // A-row within tile == output column in N-tile
  const int half = lane >> 4;

  for (int it = 0; it < RITER; ++it) {
    const int rowbase = ((blockIdx.x * 4 + wv) * RITER + it) * 16;
    for (int idx = lane; idx < 16 * K; idx += 32) {
      int r = idx / K, c = idx % K;
      As[r * KP + c] = A[(rowbase + r) * K + c];
    }
    __syncthreads();

#pragma unroll 1
    for (int nt = 0; nt < N / 16; ++nt) {
      float bv = bS[nt * 16 + m];
      v8f acc;
#pragma unroll
      for (int r = 0; r < 8; ++r) acc[r] = bv;
#pragma unroll
      for (int kc = 0; kc < K / 4; ++kc) {
        int k0 = kc * 4 + 2 * half;
        v2f a = *(const v2f*)&As[m * KP + k0];
        v2f b = *(const v2f*)&Wp[(kc * 2 + half) * (2 * N) + (nt * 16 + m) * 2];
        acc = wmma_f32_4(a, b, acc);
      }
#pragma unroll
      for (int r = 0; r < 8; ++r) {
        float v = acc[r];
        v = v > 0.f ? v : 0.f;
        out[(rowbase + r + 8 * half) * N + nt * 16 + m] = v;
      }
    }
    __syncthreads();
  }
}

// ---------------------------------------------------------------------------
// Fused RandLA conv: per target i (one wave, 16 edges):
//   rel[16,10] -> rij = relu(rel@Wpp+b)          (VALU, tiny)
//   fij = [x[src] | rij]  in LDS (16 x F, stride F+4)
//   T   = relu(fij @ Watt + batt)                (WMMA, 16xF, Watt packed LDS)
//   s   = softmax(T, axis=-1)                    (2 lanes/row, shfl_xor 16)
//   agg[i] = sum_e s*fij                         (segment sum inside wave)
// ---------------------------------------------------------------------------
template <int CIN, int WAVES, int TITER>
__global__ void randla_conv(const float* __restrict__ xin,   // [*, CIN]
                            const float* __restrict__ pos,   // [N_PTS, 3]
                            const int*   __restrict__ remap, // null => identity
                            const int*   __restrict__ idxq,  // [Nq]
                            const int*   __restrict__ src,   // [Nq*16]
                            const float* __restrict__ Wpp,   // [10, CIN]
                            const float* __restrict__ bpp,   // [CIN]
                            const float* __restrict__ Watt,  // [F, F]
                            const float* __restrict__ batt,  // [F]
                            float* __restrict__ agg) {       // [Nq, F]
  constexpr int F  = 2 * CIN;
  constexpr int FP = F + 4;
  constexpr int H  = CIN / 2;
  constexpr int NT = WAVES * 32;
  extern __shared__ float smem[];
  float* Wp    = smem;               // F*F packed
  float* WppS  = Wp + F * F;         // 10*CIN
  float* bppS  = WppS + 10 * CIN;    // CIN
  float* battS = bppS + CIN;         // F
  float* strips = battS + F;         // WAVES * 2 * 16 * FP

  const int tid  = threadIdx.x;
  const int lane = tid & 31;
  const int wv   = tid >> 5;

  for (int idx = tid; idx < F * F; idx += NT) {
    int k = idx / F, c = idx % F;
    Wp[(k >> 1) * (2 * F) + c * 2 + (k & 1)] = Watt[idx];
  }
  for (int idx = tid; idx < 10 * CIN; idx += NT) WppS[idx] = Wpp[idx];
  for (int idx = tid; idx < CIN; idx += NT) bppS[idx] = bpp[idx];
  for (int idx = tid; idx < F; idx += NT) battS[idx] = batt[idx];
  __syncthreads();

  float* fij = strips + wv * (2 * 16 * FP);
  float* T   = fij + 16 * FP;
  const int e    = lane & 15;
  const int half = lane >> 4;
  const int m    = lane & 15;

  for (int t = 0; t < TITER; ++t) {
    const int i = (blockIdx.x * WAVES + wv) * TITER + t;

    // ---- positions / relative encoding ----
    int qi = idxq[i];
    int qp = remap ? remap[qi] : qi;
    float qx = pos[qp * 3 + 0], qy = pos[qp * 3 + 1], qz = pos[qp * 3 + 2];
    int sv = src[i * KNN + e];
    int sp = remap ? remap[sv] : sv;
    float jx = pos[sp * 3 + 0], jy = pos[sp * 3 + 1], jz = pos[sp * 3 + 2];
    float vx = qx - jx, vy = qy - jy, vz = qz - jz;
    float d  = sqrtf(vx * vx + vy * vy + vz * vz);
    float rel[10] = {qx, qy, qz, jx, jy, jz, vx, vy, vz, d};

    // ---- rij = relu(rel @ Wpp + bpp); this lane covers CIN/2 features ----
#pragma unroll
    for (int f = 0; f < H; ++f) {
      int fo = half * H + f;
      float acc = bppS[fo];
#pragma unroll
      for (int c = 0; c < 10; ++c) acc += rel[c] * WppS[c * CIN + fo];
      fij[e * FP + CIN + fo] = acc > 0.f ? acc : 0.f;
    }
    // ---- gather source features ----
#pragma unroll
    for (int f = 0; f < H; ++f) {
      int fo = half * H + f;
      fij[e * FP + fo] = xin[sv * CIN + fo];
    }
    __syncthreads();

    // ---- attention GEMM: T = relu(fij[16,F] @ Watt[F,F] + batt) ----
#pragma unroll 1
    for (int nt = 0; nt < F / 16; ++nt) {
      float bv = battS[nt * 16 + m];
      v8f acc;
#pragma unroll
      for (int r = 0; r < 8; ++r) acc[r] = bv;
#pragma unroll
      for (int kc = 0; kc < F / 4; ++kc) {
        int k0 = kc * 4 + 2 * half;
        v2f a = *(const v2f*)&fij[m * FP + k0];
        v2f b = *(const v2f*)&Wp[(kc * 2 + half) * (2 * F) + (nt * 16 + m) * 2];
        acc = wmma_f32_4(a, b, acc);
      }
#pragma unroll
      for (int r = 0; r < 8; ++r) {
        float v = acc[r];
        T[(r + 8 * half) * FP + nt * 16 + m] = v > 0.f ? v : 0.f;
      }
    }
    __syncthreads();

    // ---- row softmax over F cols (lane pair L, L+16 shares row e) ----
    {
      float* row        = T   + e * FP + half * (F / 2);
      const float* frow = fij + e * FP + half * (F / 2);
      float mx = -3.4e38f;
#pragma unroll
      for (int c = 0; c < F / 2; ++c) mx = fmaxf(mx, row[c]);
      mx = fmaxf(mx, __shfl_xor(mx, 16, 32));
      float sum = 0.f;
#pragma unroll
      for (int c = 0; c < F / 2; ++c) {
        float tt = __expf(row[c] - mx);
        row[c] = tt;
        sum += tt;
      }
      sum += __shfl_xor(sum, 16, 32);
      float inv = 1.f / sum;
#pragma unroll
      for (int c = 0; c < F / 2; ++c) row[c] = row[c] * inv * frow[c];
    }
    __syncthreads();

    // ---- segment sum over the 16 edges of this target ----
#pragma unroll
    for (int cc = 0; cc < F / 32; ++cc) {
      int c = cc * 32 + lane;
      float s = 0.f;
#pragma unroll
      for (int e2 = 0; e2 < KNN; ++e2) s += T[e2 * FP + c];
      agg[i * F + c] = s;
    }
    __syncthreads();
  }
}

// ---------------------------------------------------------------------------
// Final fused kernel:
//   out = relu( relu(h2 @ W_up + b_up) + relu(x[idx1[idx2]] @ W_sc + b_sc) )
// Both weight matrices packed into LDS once per block.
// ---------------------------------------------------------------------------
__global__ void up_shortcut(const float* __restrict__ h2,   // [N2,128]
                            const float* __restrict__ x,    // [N,64]
                            const int*   __restrict__ idx1,
                            const int*   __restrict__ idx2,
                            const float* __restrict__ Wup, const float* __restrict__ bup,
                            const float* __restrict__ Wsc, const float* __restrict__ bsc,
                            float* __restrict__ out) {      // [N2,128]
  constexpr int K1 = 128, K1P = 132, K2 = 64, K2P = 68, N = 128;
  extern __shared__ float smem[];
  float* WupP = smem;                  // K1*N packed
  float* WscP = WupP + K1 * N;         // K2*N packed
  float* b1S  = WscP + K2 * N;         // N
  float* b2S  = b1S + N;               // N
  float* strips = b2S + N;             // 4 * (16*K1P + 16*K2P)

  const int tid  = threadIdx.x;
  const int lane = tid & 31;
  const int wv   = tid >> 5;

  for (int idx = tid; idx < K1 * N; idx += 128) {
    int k = idx / N, c = idx % N;
    WupP[(k >> 1) * (2 * N) + c * 2 + (k & 1)] = Wup[idx];
  }
  for (int idx = tid; idx < K2 * N; idx += 128) {
    int k = idx / N, c = idx % N;
    WscP[(k >> 1) * (2 * N) + c * 2 + (k & 1)] = Wsc[idx];
  }
  for (int idx = tid; idx < N; idx += 128) { b1S[idx] = bup[idx]; b2S[idx] = bsc[idx]; }
  __syncthreads();

  float* As = strips + wv * (16 * K1P + 16 * K2P);
  float* Xs = As + 16 * K1P;

  const int rowbase = (blockIdx.x * 4 + wv) * 16;
  for (int idx = lane; idx < 16 * K1; idx += 32) {
    int r = idx / K1, c = idx % K1;
    As[r * K1P + c] = h2[(rowbase + r) * K1 + c];
  }
  for (int idx = lane; idx < 16 * K2; idx += 32) {
    int r = idx / K2, c = idx % K2;
    int g = idx1[idx2[rowbase + r]];
    Xs[r * K2P + c] = x[g * K2 + c];
  }
  __syncthreads();

  const int m    = lane & 15;
  const int half = lane >> 4;
#pragma unroll 1
  for (int nt = 0; nt < N / 16; ++nt) {
    v8f acc1, acc2;
    float b1 = b1S[nt * 16 + m];
    float b2 = b2S[nt * 16 + m];
#pragma unroll
    for (int r = 0; r < 8; ++r) { acc1[r] = b1; acc2[r] = b2; }
#pragma unroll
    for (int kc = 0; kc < K1 / 4; ++kc) {
      int k0 = kc * 4 + 2 * half;
      v2f a = *(const v2f*)&As[m * K1P + k0];
      v2f b = *(const v2f*)&WupP[(kc * 2 + half) * (2 * N) + (nt * 16 + m) * 2];
      acc1 = wmma_f32_4(a, b, acc1);
    }
#pragma unroll
    for (int kc = 0; kc < K2 / 4; ++kc) {
      int k0 = kc * 4 + 2 * half;
      v2f a = *(const v2f*)&Xs[m * K2P + k0];
      v2f b = *(const v2f*)&WscP[(kc * 2 + half) * (2 * N) + (nt * 16 + m) * 2];
      acc2 = wmma_f32_4(a, b, acc2);
    }
#pragma unroll
    for (int r = 0; r < 8; ++r) {
      float v = fmaxf(acc1[r], 0.f) + fmaxf(acc2[r], 0.f);
      v = v > 0.f ? v : 0.f;
      out[(rowbase + r + 8 * half) * N + nt * 16 + m] = v;
    }
  }
}

// ---------------------------------------------------------------------------
extern "C" void kernel_launch(void* const* d_in, const int* in_sizes, int n_in,
                              void* d_out, int out_size, void* d_ws, size_t ws_size,
                              hipStream_t stream) {
  const float* x      = (const float*)d_in[0];
  const float* pos    = (const float*)d_in[1];
  const float* W_down = (const float*)d_in[2];
  const float* b_down = (const float*)d_in[3];
  const float* W_pp1  = (const float*)d_in[4];
  const float* b_pp1  = (const float*)d_in[5];
  const float* W_att1 = (const float*)d_in[6];
  const float* b_att1 = (const float*)d_in[7];
  const float* W_g1   = (const float*)d_in[8];
  const float* b_g1   = (const float*)d_in[9];
  const float* W_pp2  = (const float*)d_in[10];
  const float* b_pp2  = (const float*)d_in[11];
  const float* W_att2 = (const float*)d_in[12];
  const float* b_att2 = (const float*)d_in[13];
  const float* W_g2   = (const float*)d_in[14];
  const float* b_g2   = (const float*)d_in[15];
  const float* W_up   = (const float*)d_in[16];
  const float* b_up   = (const float*)d_in[17];
  const float* W_sc   = (const float*)d_in[18];
  const float* b_sc   = (const float*)d_in[19];
  const int*   idx1   = (const int*)d_in[20];
  const int*   idx2   = (const int*)d_in[21];
  const int*   src1   = (const int*)d_in[22];
  const int*   src2   = (const int*)d_in[24];
  float* out = (float*)d_out;

  // workspace carve-out (24 MB total)
  float* h0   = (float*)d_ws;                 // [N,32]
  float* agg1 = h0   + (size_t)N_PTS  * HID;  // [N1,64]
  float* h1   = agg1 + (size_t)N1_PTS * 64;   // [N1,64]
  float* agg2 = h1   + (size_t)N1_PTS * 64;   // [N2,128]
  float* h2   = agg2 + (size_t)N2_PTS * 128;  // [N2,128]

  // 1) h0 = relu(x @ W_down + b_down)            [65536,64]x[64,32]
  {
    constexpr int K = INDIM, N = HID, RI = 2;
    dim3 grid(N_PTS / (64 * RI)), block(128);
    size_t sh = (K * N + N + 4 * 16 * (K + 4)) * sizeof(float);
    gemm_bias_relu<K, N, RI><<<grid, block, sh, stream>>>(x, W_down, b_down, h0);
  }
  // 2) conv1 -> agg1                              (WMMA attention 16x64/target)
  {
    constexpr int F = 64, WV = 4, TI = 4;
    dim3 grid(N1_PTS / (WV * TI)), block(WV * 32);
    size_t sh = (F * F + 10 * 32 + 32 + F + WV * 2 * 16 * (F + 4)) * sizeof(float);
    randla_conv<32, WV, TI><<<grid, block, sh, stream>>>(h0, pos, (const int*)nullptr,
                                                         idx1, src1, W_pp1, b_pp1,
                                                         W_att1, b_att1, agg1);
  }
  // 3) h1 = relu(agg1 @ W_g1 + b_g1)              [16384,64]x[64,64]
  {
    constexpr int K = 64, N = 64, RI = 1;
    dim3 grid(N1_PTS / (64 * RI)), block(128);
    size_t sh = (K * N + N + 4 * 16 * (K + 4)) * sizeof(float);
    gemm_bias_relu<K, N, RI><<<grid, block, sh, stream>>>(agg1, W_g1, b_g1, h1);
  }
  // 4) conv2 -> agg2                              (WMMA attention 16x128/target)
  {
    constexpr int F = 128, WV = 4, TI = 4;
    dim3 grid(N2_PTS / (WV * TI)), block(WV * 32);
    size_t sh = (F * F + 10 * 64 + 64 + F + WV * 2 * 16 * (F + 4)) * sizeof(float);
    randla_conv<64, WV, TI><<<grid, block, sh, stream>>>(h1, pos, idx1,
                                                         idx2, src2, W_pp2, b_pp2,
                                                         W_att2, b_att2, agg2);
  }
  // 5) h2 = relu(agg2 @ W_g2 + b_g2)              [8192,128]x[128,128]
  {
    constexpr int K = 128, N = 128, RI = 1;
    dim3 grid(N2_PTS / (64 * RI)), block(128);
    size_t sh = (K * N + N + 4 * 16 * (K + 4)) * sizeof(float);
    gemm_bias_relu<K, N, RI><<<grid, block, sh, stream>>>(agg2, W_g2, b_g2, h2);
  }
  // 6) out = relu(relu(h2@W_up+b) + relu(x[idx1[idx2]]@W_sc+b))
  {
    dim3 grid(N2_PTS / 64), block(128);
    size_t sh = (128 * 128 + 64 * 128 + 128 + 128 + 4 * 16 * (132 + 68)) * sizeof(float);
    up_shortcut<<<grid, block, sh, stream>>>(h2, x, idx1, idx2,
                                             W_up, b_up, W_sc, b_sc, out);
  }
}